// DynamicCenterNormLoss_30133490549309
// MI455X (gfx1250) — compile-verified
//
#include <hip/hip_runtime.h>
#include <hip/hip_bf16.h>

typedef __attribute__((ext_vector_type(2))) float v2f;
typedef __attribute__((ext_vector_type(8))) float v8f;

#define NUM_CLASSES 1000
#define FEAT_DIM    512
#define BATCH       32768
#define SCALE       (1.0f / (float)BATCH)   // ND_LOSS_FACTOR == 1.0

// ---------------------------------------------------------------------------
// Kernel 1: inv_norm[l] = rsqrt(sum_d C[l,d]^2), one wave32 per class row.
// Also zeroes the scalar output (runs before the loss kernel in stream order).
// ---------------------------------------------------------------------------
__global__ void center_invnorm_kernel(const float* __restrict__ C,
                                      float* __restrict__ inv_norm,
                                      float* __restrict__ out) {
  if (blockIdx.x == 0 && threadIdx.x == 0) out[0] = 0.0f;

  const int wave = (int)((blockIdx.x * blockDim.x + threadIdx.x) >> 5);
  const int lane = (int)(threadIdx.x & 31);
  if (wave >= NUM_CLASSES) return;

  const float* row = C + (size_t)wave * FEAT_DIM;
  float s = 0.0f;
#pragma unroll
  for (int j = 0; j < FEAT_DIM / 32; ++j) {        // coalesced: stride-32 lanes
    float v = row[j * 32 + lane];
    s += v * v;
  }
#pragma unroll
  for (int off = 16; off > 0; off >>= 1)
    s += __shfl_xor(s, off, 32);
  if (lane == 0) inv_norm[wave] = rsqrtf(s);
}

// ---------------------------------------------------------------------------
// Kernel 2: one wave32 per 16-row tile.
//   A (16x4 f32): lane l<16 row=l holds K = k0,k0+1 ; lane l+16 holds k0+2,k0+3
//   B (4x16 f32): lane n<16 col=n holds K = k0,k0+1 ; lane n+16 holds k0+2,k0+3
//   -> both operands load one float2 at (rowbase + kk + 2*half) per step.
//   Diagonal of the 16x16 accumulator = per-row dot(s_row, C[label_row]).
//   s^2 / t^2 norms accumulate on the same streamed data.
// ---------------------------------------------------------------------------
__global__ void nd_loss_kernel(const float* __restrict__ S,
                               const float* __restrict__ T,
                               const int*   __restrict__ labels,
                               const float* __restrict__ C,
                               const float* __restrict__ inv_norm,
                               float* __restrict__ out) {
  const int lane = (int)(threadIdx.x & 31);
  const int half = lane >> 4;          // 0: K lo pair, 1: K hi pair
  const int lm   = lane & 15;          // row (A) / col (B) within tile
  const int tile = (int)(blockIdx.x * (blockDim.x >> 5) + (threadIdx.x >> 5));
  const int row  = tile * 16 + lm;
  const int lbl  = labels[row];

  const float* sp = S + (size_t)row * FEAT_DIM + (size_t)(half * 2);
  const float* tp = T + (size_t)row * FEAT_DIM + (size_t)(half * 2);
  const float* cp = C + (size_t)lbl * FEAT_DIM + (size_t)(half * 2);

  v8f acc = {};
  float s2 = 0.0f, t2 = 0.0f;

#pragma unroll 4
  for (int kk = 0; kk < FEAT_DIM; kk += 4) {
    v2f a  = *(const v2f*)(sp + kk);   // A: s rows
    v2f b  = *(const v2f*)(cp + kk);   // B: center rows (L2-resident)
    v2f tv = *(const v2f*)(tp + kk);   // ride-along for t-norm
    s2 += a.x * a.x + a.y * a.y;
    t2 += tv.x * tv.x + tv.y * tv.y;
    // D += A x B ; diagonal D[i][i] = dot(s_row_i, C[label_i]) (partial in K)
    acc = __builtin_amdgcn_wmma_f32_16x16x4_f32(
        /*neg_a=*/false, a, /*neg_b=*/false, b,
        /*c_mod=*/(short)0, acc, /*reuse_a=*/false, /*reuse_b=*/false);
  }

  // --- extract diagonal: D[i][i] is acc[i&7] on lane i (i<8) / lane 16+i (i>=8)
  const int idx = lm & 7;
  float dsel = acc[0];
#pragma unroll
  for (int i = 1; i < 8; ++i) dsel = (idx == i) ? acc[i] : dsel;
  const float dup  = __shfl_xor(dsel, 16, 32);
  const float diag = (lm < 8) ? dsel : dup;

  // --- per-row squared norms: combine the two K-halves
  const float s2row = s2 + __shfl_xor(s2, 16, 32);
  const float t2row = t2 + __shfl_xor(t2, 16, 32);

  const float invn = inv_norm[lbl];
  const float mx   = fmaxf(s2row, t2row);          // max norm^2; sqrt is monotone
  float contrib = (half == 0) ? (1.0f - diag * invn * rsqrtf(mx)) : 0.0f;

#pragma unroll
  for (int off = 16; off > 0; off >>= 1)
    contrib += __shfl_xor(contrib, off, 32);
  if (lane == 0) atomicAdd(out, contrib * SCALE);
}

// ---------------------------------------------------------------------------
extern "C" void kernel_launch(void* const* d_in, const int* in_sizes, int n_in,
                              void* d_out, int out_size, void* d_ws, size_t ws_size,
                              hipStream_t stream) {
  const float* s_emb   = (const float*)d_in[0];
  const float* t_emb   = (const float*)d_in[1];
  const int*   labels  = (const int*)d_in[2];
  const float* centers = (const float*)d_in[3];
  float*       out     = (float*)d_out;
  float*       invn    = (float*)d_ws;            // NUM_CLASSES floats

  // 1000 waves -> 125 blocks of 8 waves
  center_invnorm_kernel<<<NUM_CLASSES / 8, 256, 0, stream>>>(centers, invn, out);
  // 2048 tiles of 16 rows -> 256 blocks of 8 waves
  nd_loss_kernel<<<(BATCH / 16) / 8, 256, 0, stream>>>(s_emb, t_emb, labels,
                                                       centers, invn, out);
}